// SymmetricChannel_22445499089175
// MI455X (gfx1250) — compile-verified
//
#include <hip/hip_runtime.h>
#include <stdint.h>

// SymmetricChannel forward: memory-bound streaming kernel.
// One workgroup (256 threads = 8 wave32) per row of 1024 floats.
// messages row is staged into LDS with CDNA5 async global->LDS copies
// (overlaps with the normal-VMEM probs/mask stream); outputs are written
// with non-temporal stores (pure streaming, ~0.92 GB total traffic,
// ~40us at 23.3 TB/s HBM).

typedef float f4 __attribute__((ext_vector_type(4)));

#define VSZ   1024
#define TAILN 1023
#define PERRC 0.1f

__global__ __launch_bounds__(256) void symchan_kernel(
    const float* __restrict__ messages,
    const float* __restrict__ probs,
    const int*   __restrict__ mask,
    float* __restrict__ out,
    long long plane)
{
    __shared__ __attribute__((aligned(16))) float smem[VSZ];  // staged messages row
    __shared__ __attribute__((aligned(16))) float red[8];     // wave partial sums

    const int row = blockIdx.x;    // 0 .. 32767
    const int t   = threadIdx.x;   // 0 .. 255
    const int c0  = t << 2;        // 4 columns per thread

    const float* mrow = messages + (long long)row * VSZ;
    const float* prow = probs    + (long long)row * VSZ;
    const int*   krow = mask     + (long long)row * TAILN;

    // ---- CDNA5 async copy: stage this thread's 16B of the messages row into
    // ---- LDS (ASYNCcnt path) while the normal VMEM path streams probs/mask.
    {
        unsigned lds_off = (unsigned)(uintptr_t)(&smem[c0]);  // low 32b of generic ptr == LDS offset
        const float* gsrc = mrow + c0;
        asm volatile("global_load_async_to_lds_b128 %0, %1, off"
                     :
                     : "v"(lds_off), "v"(gsrc)
                     : "memory");
    }

    // Overlapped with the async transfer:
    f4    p  = *(const f4*)(prow + c0);
    float p0 = prow[0];  // block-uniform -> scalar (SMEM) load

    float mk[4];
#pragma unroll
    for (int i = 0; i < 4; ++i) {
        int   c  = c0 + i;
        int   mi = c ? (c - 1) : 0;      // clamp avoids OOB read at c==0
        float v  = (float)krow[mi];
        mk[i]    = c ? v : 0.0f;         // column 0 (EOS) carries no mask
    }

    // Wait for this wave's async transfer, then pull the tile from LDS.
    asm volatile("s_wait_asynccnt 0" ::: "memory");
    f4 m = *(const f4*)(&smem[c0]);

    const float inv = 1.0f / (float)(VSZ - 2);

    float T[4];
#pragma unroll
    for (int i = 0; i < 4; ++i) T[i] = m[i] * mk[i];

    // ---- row_sum reduction: wave32 shuffle tree, then 8 wave partials in LDS
    float partial = (T[0] + T[1]) + (T[2] + T[3]);
#pragma unroll
    for (int off = 16; off > 0; off >>= 1)
        partial += __shfl_down(partial, off, 32);
    if ((t & 31) == 0) red[t >> 5] = partial;
    __syncthreads();
    f4 r0 = *(const f4*)(&red[0]);
    f4 r1 = *(const f4*)(&red[4]);
    const float rs = ((r0.x + r0.y) + (r0.z + r0.w)) +
                     ((r1.x + r1.y) + (r1.z + r1.w));

    // ---- elementwise outputs
    const float ip1  = 1.0f + inv;
    const float keep = 1.0f - PERRC;
    const float rep  = PERRC * inv;

    f4 nm, np;
#pragma unroll
    for (int i = 0; i < 4; ++i) {
        int   c   = c0 + i;
        float mv  = m[i];
        float pv  = p[i];
        float nmv = mv + rs * inv - T[i] * ip1;               // noisy message tail
        float npv = pv * keep + (1.0f - pv - p0) * rep;       // noisy prob tail
        nm[i] = c ? nmv : mv;                                 // col 0 passthrough
        np[i] = c ? npv : pv;
    }

    // ---- streaming (non-temporal) stores: nothing below is re-read
    const long long o = (long long)row * VSZ + c0;
    __builtin_nontemporal_store(nm, (f4*)(out + o));                 // noisy_messages
    __builtin_nontemporal_store(np, (f4*)(out + plane + o));         // noisy_probs
    __builtin_nontemporal_store(m,  (f4*)(out + 2 * plane + o));     // clean messages
    __builtin_nontemporal_store(p,  (f4*)(out + 3 * plane + o));     // clean probs
}

extern "C" void kernel_launch(void* const* d_in, const int* in_sizes, int n_in,
                              void* d_out, int out_size, void* d_ws, size_t ws_size,
                              hipStream_t stream) {
    const float* messages = (const float*)d_in[0];
    const float* probs    = (const float*)d_in[1];
    const int*   mask     = (const int*)d_in[2];   // bernoulli mask, integer layout
    float* out = (float*)d_out;

    const int rows = in_sizes[0] / VSZ;                 // B*L = 32768
    const long long plane = (long long)rows * VSZ;      // 33,554,432 elements/plane

    symchan_kernel<<<dim3(rows), dim3(256), 0, stream>>>(messages, probs, mask, out, plane);

    (void)n_in; (void)out_size; (void)d_ws; (void)ws_size;
}